// CopyDecoder_34557306864190
// MI455X (gfx1250) — compile-verified
//
#include <hip/hip_runtime.h>
#include <hip/hip_bf16.h>

#define B 64
#define S 200
#define V 50257
#define H 512
#define E 512
#define XDIM 1536   // E + 2H
#define G3H 1536    // 3*H

typedef __attribute__((ext_vector_type(2))) float v2f;
typedef __attribute__((ext_vector_type(8))) float v8f;
typedef __attribute__((ext_vector_type(4))) unsigned int u32x4;
typedef __attribute__((ext_vector_type(8))) int i32x8;
typedef __attribute__((ext_vector_type(4))) int i32x4;

#if defined(__gfx1250__) && defined(__has_builtin)
#if __has_builtin(__builtin_amdgcn_tensor_load_to_lds) && __has_builtin(__builtin_amdgcn_s_wait_tensorcnt)
#define USE_TDM 1
#endif
#endif
#ifndef USE_TDM
#define USE_TDM 0
#endif

// ---------------------------------------------------------------------------
// K0a: x = concat(embed[input_idx], order ? weighted : 0)   -> (B, 1536)
// ---------------------------------------------------------------------------
__global__ void k_build_x(const int* __restrict__ input_idx,
                          const float* __restrict__ embed,
                          const float* __restrict__ weighted,
                          const int* __restrict__ order,
                          float* __restrict__ x) {
    int i = blockIdx.x * blockDim.x + threadIdx.x;
    if (i >= B * XDIM) return;
    int b = i / XDIM, c = i % XDIM;
    float v;
    if (c < E) {
        v = embed[(size_t)input_idx[b] * E + c];
    } else {
        v = (order[0] != 0) ? weighted[b * (2 * H) + (c - E)] : 0.0f;
    }
    x[i] = v;
}

// ---------------------------------------------------------------------------
// K0b: h0 = order ? prev_state : encoded[:, -1] @ Ws_w.T + Ws_b   -> (B, 512)
// ---------------------------------------------------------------------------
__global__ void k_build_h0(const float* __restrict__ prev_state,
                           const float* __restrict__ encoded,
                           const float* __restrict__ Ws_w,
                           const float* __restrict__ Ws_b,
                           const int* __restrict__ order,
                           float* __restrict__ h0) {
    int i = blockIdx.x * blockDim.x + threadIdx.x;
    if (i >= B * H) return;
    if (order[0] != 0) { h0[i] = prev_state[i]; return; }
    int b = i >> 9, h = i & (H - 1);
    const float* er = encoded + ((size_t)b * S + (S - 1)) * (2 * H);
    const float* wr = Ws_w + (size_t)h * (2 * H);
    float acc = Ws_b[h];
    for (int k = 0; k < 2 * H; ++k) acc += er[k] * wr[k];
    h0[i] = acc;
}

// ---------------------------------------------------------------------------
// K1: gi = x @ W_ih.T  (K=1536),  gh = h0 @ W_hh.T (K=512)
// One wave per 16x16 output tile; WMMA f32 16x16x4.
// grid = (96 ntiles, 4 mtiles, 2 which)
// ---------------------------------------------------------------------------
__global__ __launch_bounds__(32) void k_gru_gemm(const float* __restrict__ x,
                                                 const float* __restrict__ h0,
                                                 const float* __restrict__ W_ih,
                                                 const float* __restrict__ W_hh,
                                                 float* __restrict__ gi,
                                                 float* __restrict__ gh) {
    const int nt = blockIdx.x;
    const int mt = blockIdx.y;
    const int which = blockIdx.z;
    const float* A = which ? h0 : x;
    const float* W = which ? W_hh : W_ih;
    float* Out = which ? gh : gi;
    const int K = which ? H : XDIM;

    const int lane = threadIdx.x;
    const int mrow = mt * 16 + (lane & 15);
    const int ncol = nt * 16 + (lane & 15);
    const int khalf = (lane >> 4) * 2;

    v8f acc = {};
    for (int k0 = 0; k0 < K; k0 += 4) {
        v2f a = *(const v2f*)(A + (size_t)mrow * K + k0 + khalf);
        v2f bm = *(const v2f*)(W + (size_t)ncol * K + k0 + khalf);
        acc = __builtin_amdgcn_wmma_f32_16x16x4_f32(false, a, false, bm,
                                                    (short)0, acc, false, false);
    }
    const int rbase = mt * 16 + ((lane >> 4) ? 8 : 0);
    const int c = nt * 16 + (lane & 15);
#pragma unroll
    for (int j = 0; j < 8; ++j)
        Out[(size_t)(rbase + j) * G3H + c] = acc[j];
}

// ---------------------------------------------------------------------------
// K2: GRU gate nonlinearity -> state  (B, 512)
// ---------------------------------------------------------------------------
__global__ void k_gates(const float* __restrict__ gi, const float* __restrict__ gh,
                        const float* __restrict__ b_ih, const float* __restrict__ b_hh,
                        const float* __restrict__ h0,
                        float* __restrict__ state, float* __restrict__ out_state) {
    int i = blockIdx.x * blockDim.x + threadIdx.x;
    if (i >= B * H) return;
    int b = i >> 9, h = i & (H - 1);
    const float* gib = gi + (size_t)b * G3H;
    const float* ghb = gh + (size_t)b * G3H;
    float ir = gib[h]          + b_ih[h];
    float iz = gib[H + h]      + b_ih[H + h];
    float in_ = gib[2 * H + h] + b_ih[2 * H + h];
    float hr = ghb[h]          + b_hh[h];
    float hz = ghb[H + h]      + b_hh[H + h];
    float hn = ghb[2 * H + h]  + b_hh[2 * H + h];
    float r = 1.0f / (1.0f + __expf(-(ir + hr)));
    float z = 1.0f / (1.0f + __expf(-(iz + hz)));
    float n = tanhf(in_ + r * hn);
    float s = (1.0f - z) * n + z * h0[i];
    state[i] = s;
    out_state[i] = s;
}

// ---------------------------------------------------------------------------
// K3: score_g = state @ Wo_w.T + Wo_b   (64 x 50257)
// Block = 128 threads (4 waves = 4 consecutive N-tiles). 64x64 Wo block is
// staged in LDS with coalesced float4 loads; row stride padded to 68 dwords
// so the WMMA ds_load_b64 pattern is bank-conflict-free. Each wave holds 4
// M-tile accumulators so every Wo element is loaded from HBM exactly once.
// grid.x = ceil(V/64) = 786
// ---------------------------------------------------------------------------
#define WSTRIDE 68
__global__ __launch_bounds__(128) void k_score_g(const float* __restrict__ state,
                                                 const float* __restrict__ Wo_w,
                                                 const float* __restrict__ Wo_b,
                                                 float* __restrict__ scoreg) {
    __shared__ float ldsW[64 * WSTRIDE];    // ~17 KB
    const int tid = threadIdx.x;
    const int wave = tid >> 5;
    const int lane = tid & 31;
    const int n0 = blockIdx.x * 64;
    const int nIdx = n0 + wave * 16 + (lane & 15);
    const int khalf = (lane >> 4) * 2;
    const int m = lane & 15;

    v8f acc0 = {}, acc1 = {}, acc2 = {}, acc3 = {};
    for (int ks = 0; ks < H; ks += 64) {
        __syncthreads();
        for (int i = tid; i < 64 * 16; i += 128) {       // 1024 float4
            int row = i >> 4;
            int col = (i & 15) * 4;
            int rC = n0 + row; if (rC > V - 1) rC = V - 1;
            float4 wv = *(const float4*)(Wo_w + (size_t)rC * H + ks + col);
            *(float4*)(ldsW + row * WSTRIDE + col) = wv;
        }
        __syncthreads();
        const float* lw = ldsW + (wave * 16 + (lane & 15)) * WSTRIDE;
        for (int kk = 0; kk < 64; kk += 4) {
            v2f bm = *(const v2f*)(lw + kk + khalf);
            const int kg = ks + kk + khalf;
            v2f a0 = *(const v2f*)(state + (size_t)(m)      * H + kg);
            v2f a1 = *(const v2f*)(state + (size_t)(16 + m) * H + kg);
            v2f a2 = *(const v2f*)(state + (size_t)(32 + m) * H + kg);
            v2f a3 = *(const v2f*)(state + (size_t)(48 + m) * H + kg);
            acc0 = __builtin_amdgcn_wmma_f32_16x16x4_f32(false, a0, false, bm, (short)0, acc0, false, false);
            acc1 = __builtin_amdgcn_wmma_f32_16x16x4_f32(false, a1, false, bm, (short)0, acc1, false, false);
            acc2 = __builtin_amdgcn_wmma_f32_16x16x4_f32(false, a2, false, bm, (short)0, acc2, false, false);
            acc3 = __builtin_amdgcn_wmma_f32_16x16x4_f32(false, a3, false, bm, (short)0, acc3, false, false);
        }
    }
    if (nIdx < V) {
        const float bias = Wo_b[nIdx];
        const int rb = (lane >> 4) ? 8 : 0;
#pragma unroll
        for (int j = 0; j < 8; ++j) {
            scoreg[(size_t)(rb + j)      * V + nIdx] = acc0[j] + bias;
            scoreg[(size_t)(16 + rb + j) * V + nIdx] = acc1[j] + bias;
            scoreg[(size_t)(32 + rb + j) * V + nIdx] = acc2[j] + bias;
            scoreg[(size_t)(48 + rb + j) * V + nIdx] = acc3[j] + bias;
        }
    }
}

// ---------------------------------------------------------------------------
// K4: score_c[b][s] = tanh( sum_h tanh(encoded[b,s]@Wc[h]+Wc_b[h])*state[b,h]
//                           + padmask )
// Block = 128 threads (4 waves). Encoded tile (16 x 512 f32, half of K) is
// DMA'd into LDS by the Tensor Data Mover with pad_interval=256dw/
// pad_amount=2dw -> padded row stride 516 dwords, bank-conflict-free for the
// WMMA A-pattern ds_load_b64. Two K-halves accumulate into 8 persistent
// accumulators per wave. grid = (B, ceil(S/16)=13)
// ---------------------------------------------------------------------------
#define TROWS 16
#define TK 512
#define TPAD ((TROWS * TK) + (((TROWS * TK) >> 8) << 1))   // 8256 dwords

__global__ __launch_bounds__(128) void k_score_c(const float* __restrict__ encoded,
                                                 const float* __restrict__ Wc_w,
                                                 const float* __restrict__ Wc_b,
                                                 const float* __restrict__ state,
                                                 const int* __restrict__ encoded_idx,
                                                 float* __restrict__ scorec) {
    __shared__ float tile[TPAD];            // ~33 KB
    const int b = blockIdx.x;
    const int st = blockIdx.y;
    const int tid = threadIdx.x;
    const int wave = tid >> 5;
    const int lane = tid & 31;
    const int m = lane & 15;
    const int khalf = (lane >> 4) * 2;

    v8f acc[8];
#pragma unroll
    for (int t = 0; t < 8; ++t) acc[t] = (v8f){};

    for (int half = 0; half < 2; ++half) {
        const int kh = half * TK;
        __syncthreads();
#if USE_TDM
        if (wave == 0) {
            // ---- Tensor DMA descriptor (CDNA5 ISA ch.8) ----
            const uint64_t ga = (uint64_t)(uintptr_t)(encoded +
                                ((size_t)b * S + st * TROWS) * (2 * H) + kh);
            const uint32_t ldsb = (uint32_t)(uintptr_t)tile;   // LDS byte offset
            const uint32_t dim1 = (uint32_t)(S - st * TROWS);  // rows left; OOB rows -> 0

            u32x4 g0;
            g0[0] = 1u;                                         // count=1, user mode
            g0[1] = ldsb;                                       // lds_addr
            g0[2] = (uint32_t)ga;                               // global_addr[31:0]
            g0[3] = (uint32_t)((ga >> 32) & 0x1FFFFFFu)         // global_addr[56:32]
                    | (2u << 30);                               // type=2 (image)

            i32x8 g1;
            g1[0] = (int)((2u << 16)        // data_size = 4B
                    | (1u << 20)            // pad_enable
                    | (7u << 22)            // pad_interval: every 256 dwords
                    | (1u << 25));          // pad_amount: 2 dwords
            g1[1] = (int)((TK & 0xFFFFu) << 16);                // tensor_dim0[15:0]
            g1[2] = (int)(((TK >> 16) & 0xFFFFu) | (dim1 << 16)); // dim0 hi | tensor_dim1 lo
            g1[3] = (int)(((dim1 >> 16) & 0xFFFFu) | ((uint32_t)TK << 16)); // dim1 hi | tile_dim0
            g1[4] = TROWS;                                      // tile_dim1 (tile_dim2=0)
            g1[5] = 2 * H;                                      // tensor_dim0_stride (1024)
            g1[6] = 0;
            g1[7] = 0;

            i32x4 gz4 = {};
            i32x8 gz8 = {};
            __builtin_amdgcn_tensor_load_to_lds(g0, g1, gz4, gz4, gz8, 0);
            __builtin_amdgcn_s_wait_tensorcnt(0);
        }
#else
        for (int i = tid; i < TROWS * TK / 4; i += 128) {
            int r = i >> 7;                 // 128 float4 per row
            int c = (i & 127) * 4;
            int s = st * TROWS + r;
            float4 vv;
            if (s < S) vv = *(const float4*)(encoded + ((size_t)b * S + s) * (2 * H) + kh + c);
            else       vv = make_float4(0.f, 0.f, 0.f, 0.f);
            int d = r * TK + c;
            int ad = d + ((d >> 8) << 1);   // padded layout: +2dw per 256dw
            *(v2f*)(tile + ad)     = (v2f){vv.x, vv.y};
            *(v2f*)(tile + ad + 2) = (v2f){vv.z, vv.w};
        }
#endif
        __syncthreads();

#pragma unroll
        for (int t = 0; t < 8; ++t) {
            const int h = (wave * 8 + t) * 16 + (lane & 15);
            const float* wrow = Wc_w + (size_t)h * (2 * H) + kh;
            v8f a_ = acc[t];
            for (int k0 = 0; k0 < TK; k0 += 4) {
                const int d = m * TK + k0 + khalf;
                v2f av = *(const v2f*)(tile + d + ((d >> 8) << 1));
                v2f bv = *(const v2f*)(wrow + k0 + khalf);
                a_ = __builtin_amdgcn_wmma_f32_16x16x4_f32(false, av, false, bv,
                                                           (short)0, a_, false, false);
            }
            acc[t] = a_;
        }
    }

    // epilogue: tanh(gemm + bias) * state[h], reduce over h
    float partial[8];
#pragma unroll
    for (int j = 0; j < 8; ++j) partial[j] = 0.0f;
#pragma unroll
    for (int t = 0; t < 8; ++t) {
        const int h = (wave * 8 + t) * 16 + (lane & 15);
        const float sv = state[b * H + h];
        const float bias = Wc_b[h];
#pragma unroll
        for (int j = 0; j < 8; ++j)
            partial[j] += tanhf(acc[t][j] + bias) * sv;
    }
#pragma unroll
    for (int j = 0; j < 8; ++j) {
        float p = partial[j];
        p += __shfl_xor(p, 1, 32);
        p += __shfl_xor(p, 2, 32);
        p += __shfl_xor(p, 4, 32);
        p += __shfl_xor(p, 8, 32);
        partial[j] = p;
    }
    __syncthreads();                        // tile reads done; reuse as scratch
    float* red = tile;
    if ((lane & 15) == 0) {
        const int srow = (lane >> 4) * 8;
#pragma unroll
        for (int j = 0; j < 8; ++j) red[wave * 16 + srow + j] = partial[j];
    }
    __syncthreads();
    if (tid < 16) {
        float v = red[tid] + red[16 + tid] + red[32 + tid] + red[48 + tid];
        int s = st * TROWS + tid;
        if (s < S) {
            float pad = (encoded_idx[b * S + s] == 0) ? -1000.0f : 0.0f;
            scorec[b * S + s] = tanhf(v + pad);
        }
    }
}

// ---------------------------------------------------------------------------
// K5: softmax over [score_g | score_c], prob_g out, scatter-add prob_c,
// attention + weighted_new. One 256-thread block per batch row.
// ---------------------------------------------------------------------------
__global__ __launch_bounds__(256) void k_softmax_out(const float* __restrict__ scoreg,
                                                     const float* __restrict__ scorec,
                                                     const int* __restrict__ encoded_idx,
                                                     const int* __restrict__ input_idx,
                                                     const float* __restrict__ encoded,
                                                     float* __restrict__ out_probs,
                                                     float* __restrict__ out_weighted) {
    __shared__ float red[256];
    __shared__ float pc[S];
    __shared__ float attn[S];
    const int b = blockIdx.x;
    const int tid = threadIdx.x;
    const float* sg = scoreg + (size_t)b * V;
    const float* sc = scorec + b * S;
    const int* eidx = encoded_idx + b * S;

    float mx = -1e30f;
    for (int i = tid; i < V; i += 256) mx = fmaxf(mx, sg[i]);
    if (tid < S) mx = fmaxf(mx, sc[tid]);
    red[tid] = mx; __syncthreads();
    for (int s = 128; s > 0; s >>= 1) { if (tid < s) red[tid] = fmaxf(red[tid], red[tid + s]); __syncthreads(); }
    mx = red[0]; __syncthreads();

    float sum = 0.0f;
    for (int i = tid; i < V; i += 256) sum += __expf(sg[i] - mx);
    if (tid < S) { float e = __expf(sc[tid] - mx); pc[tid] = e; sum += e; }
    red[tid] = sum; __syncthreads();
    for (int s = 128; s > 0; s >>= 1) { if (tid < s) red[tid] += red[tid + s]; __syncthreads(); }
    const float inv = 1.0f / red[0]; __syncthreads();

    float* ob = out_probs + (size_t)b * V;
    for (int i = tid; i < V; i += 256) ob[i] = __expf(sg[i] - mx) * inv;
    if (tid < S) pc[tid] *= inv;
    __threadfence();
    __syncthreads();

    if (tid < S) atomicAdd(&ob[eidx[tid]], pc[tid]);

    const int ii = input_idx[b];
    float match = (tid < S && eidx[tid] == ii) ? 1.0f : 0.0f;
    red[tid] = match; __syncthreads();
    for (int s = 128; s > 0; s >>= 1) { if (tid < s) red[tid] += red[tid + s]; __syncthreads(); }
    const float cnt = red[0];
    const float denom = (cnt > 1.0f) ? cnt : 1.0f;
    if (tid < S) attn[tid] = pc[tid] * match / denom;
    __syncthreads();

    const float* eb = encoded + (size_t)b * S * (2 * H);
    float* wb = out_weighted + b * (2 * H);
    for (int e = tid; e < 2 * H; e += 256) {
        float acc = 0.0f;
        for (int s = 0; s < S; ++s) acc += attn[s] * eb[(size_t)s * (2 * H) + e];
        wb[e] = acc;
    }
}

// ---------------------------------------------------------------------------
extern "C" void kernel_launch(void* const* d_in, const int* in_sizes, int n_in,
                              void* d_out, int out_size, void* d_ws, size_t ws_size,
                              hipStream_t stream) {
    const int*   input_idx   = (const int*)d_in[0];
    const float* encoded     = (const float*)d_in[1];
    const int*   encoded_idx = (const int*)d_in[2];
    const float* prev_state  = (const float*)d_in[3];
    const float* weighted    = (const float*)d_in[4];
    const int*   order       = (const int*)d_in[5];
    const float* embed       = (const float*)d_in[6];
    const float* W_ih        = (const float*)d_in[7];
    const float* W_hh        = (const float*)d_in[8];
    const float* b_ih        = (const float*)d_in[9];
    const float* b_hh        = (const float*)d_in[10];
    const float* Ws_w        = (const float*)d_in[11];
    const float* Ws_b        = (const float*)d_in[12];
    const float* Wo_w        = (const float*)d_in[13];
    const float* Wo_b        = (const float*)d_in[14];
    const float* Wc_w        = (const float*)d_in[15];
    const float* Wc_b        = (const float*)d_in[16];

    float* ws = (float*)d_ws;
    float* x      = ws;                       // 64*1536
    float* h0     = x + B * XDIM;             // 64*512
    float* gi     = h0 + B * H;               // 64*1536
    float* gh     = gi + B * G3H;             // 64*1536
    float* state  = gh + B * G3H;             // 64*512
    float* scorec = state + B * H;            // 64*200
    float* scoreg = scorec + B * S;           // 64*50257

    float* out_probs    = (float*)d_out;
    float* out_state    = out_probs + (size_t)B * V;
    float* out_weighted = out_state + (size_t)B * H;

    k_build_x<<<(B * XDIM + 255) / 256, 256, 0, stream>>>(input_idx, embed, weighted, order, x);
    k_build_h0<<<(B * H + 255) / 256, 256, 0, stream>>>(prev_state, encoded, Ws_w, Ws_b, order, h0);

    k_gru_gemm<<<dim3(G3H / 16, B / 16, 2), 32, 0, stream>>>(x, h0, W_ih, W_hh, gi, gh);

    k_gates<<<(B * H + 255) / 256, 256, 0, stream>>>(gi, gh, b_ih, b_hh, h0, state, out_state);

    k_score_g<<<(V + 63) / 64, 128, 0, stream>>>(state, Wo_w, Wo_b, scoreg);

    k_score_c<<<dim3(B, (S + 15) / 16), 128, 0, stream>>>(encoded, Wc_w, Wc_b, state, encoded_idx, scorec);

    k_softmax_out<<<B, 256, 0, stream>>>(scoreg, scorec, encoded_idx, input_idx, encoded,
                                         out_probs, out_weighted);
}